// myMSEloss_33973191311485
// MI455X (gfx1250) — compile-verified
//
#include <hip/hip_runtime.h>

typedef __attribute__((ext_vector_type(2))) float v2f;
typedef __attribute__((ext_vector_type(8))) float v8f;

// Sum a float across all 32 lanes of a wave using the CDNA5 matrix pipe.
// A (16x4 f32, 2 VGPRs): VGPR0 = per-lane partial, VGPR1 = 0.
//   => A[m][0] = p_m (lanes 0-15), A[m][2] = p_{m+16} (lanes 16-31), A[m][1]=A[m][3]=0
// B (4x16) = all ones  =>  D[m][n] = p_m + p_{m+16}  for every column n.
// Each lane sums its 8 D VGPRs: lanes 0-15 get sum over rows 0-7,
// lanes 16-31 get sum over rows 8-15; one shfl_xor(16) completes the total.
// EXEC is all-1s at every call site (no divergence there), as WMMA requires.
__device__ __forceinline__ float wave_sum32(float v) {
  v2f a; a[0] = v;    a[1] = 0.0f;
  v2f b; b[0] = 1.0f; b[1] = 1.0f;
  v8f c = {};
  v8f d = __builtin_amdgcn_wmma_f32_16x16x4_f32(
      /*neg_a=*/false, a, /*neg_b=*/false, b,
      /*c_mod=*/(short)0, c, /*reuse_a=*/false, /*reuse_b=*/false);
  float r = ((d[0] + d[1]) + (d[2] + d[3])) + ((d[4] + d[5]) + (d[6] + d[7]));
  r += __shfl_xor(r, 16, 32);
  return r;
}

// Per-row weighted squared error. Class c = l[0] (exact small float).
// w0 = 10 - 3c ; w_k = (k <= 3c) ? 1 : 0 for k in 1..9  -> branchless cndmask.
__device__ __forceinline__ float row_loss(const float o[10], const float l[10]) {
  const float fl   = l[0];               // class id as float (exact)
  const float w0   = 10.0f - 3.0f * fl;
  const float flim = 3.0f * fl;          // exact: 0, 3, 6, 9
  float d0 = o[0] - l[0];
  float s  = w0 * (d0 * d0);
#pragma unroll
  for (int k = 1; k < 10; ++k) {
    float dk = o[k] - l[k];
    float t  = dk * dk;
    s += ((float)k <= flim) ? t : 0.0f;
  }
  return s;
}

__global__ __launch_bounds__(256) void wloss_partial(
    const float* __restrict__ outputs, const float* __restrict__ labels,
    float* __restrict__ partials, long long npairs, long long oddRow) {
  const float4* __restrict__ O = reinterpret_cast<const float4*>(outputs);
  const float4* __restrict__ L = reinterpret_cast<const float4*>(labels);

  float acc = 0.0f;
  const long long stride = (long long)gridDim.x * blockDim.x;
  for (long long p = (long long)blockIdx.x * blockDim.x + threadIdx.x;
       p < npairs; p += stride) {
    const long long q = 5 * p;          // 2 rows = 80 bytes = 5 x b128, 16B aligned
    float4 o0 = O[q + 0], o1 = O[q + 1], o2 = O[q + 2], o3 = O[q + 3], o4 = O[q + 4];
    float4 l0 = L[q + 0], l1 = L[q + 1], l2 = L[q + 2], l3 = L[q + 3], l4 = L[q + 4];

    float oa[10] = {o0.x, o0.y, o0.z, o0.w, o1.x, o1.y, o1.z, o1.w, o2.x, o2.y};
    float la[10] = {l0.x, l0.y, l0.z, l0.w, l1.x, l1.y, l1.z, l1.w, l2.x, l2.y};
    float ob[10] = {o2.z, o2.w, o3.x, o3.y, o3.z, o3.w, o4.x, o4.y, o4.z, o4.w};
    float lb[10] = {l2.z, l2.w, l3.x, l3.y, l3.z, l3.w, l4.x, l4.y, l4.z, l4.w};

    acc += row_loss(oa, la);
    acc += row_loss(ob, lb);
  }

  // Generic tail for odd row counts (not hit for B = 8388608).
  if (oddRow >= 0 && blockIdx.x == 0 && threadIdx.x == 0) {
    const float* o = outputs + oddRow * 10;
    const float* l = labels  + oddRow * 10;
    float oo[10], ll[10];
#pragma unroll
    for (int k = 0; k < 10; ++k) { oo[k] = o[k]; ll[k] = l[k]; }
    acc += row_loss(oo, ll);
  }

  // Wave reduce on the matrix pipe, then combine the block's 8 waves via LDS.
  float wsum = wave_sum32(acc);
  __shared__ float waves[8];
  const int lane = threadIdx.x & 31;
  const int wid  = threadIdx.x >> 5;
  if (lane == 0) waves[wid] = wsum;
  __syncthreads();
  if (threadIdx.x == 0) {
    float b = 0.0f;
#pragma unroll
    for (int i = 0; i < 8; ++i) b += waves[i];
    partials[blockIdx.x] = b;
  }
}

__global__ __launch_bounds__(256) void wloss_final(
    const float* __restrict__ partials, int n, float* __restrict__ out, float invB) {
  float acc = 0.0f;
  for (int i = threadIdx.x; i < n; i += 256) acc += partials[i];
  float wsum = wave_sum32(acc);
  __shared__ float waves[8];
  const int lane = threadIdx.x & 31;
  const int wid  = threadIdx.x >> 5;
  if (lane == 0) waves[wid] = wsum;
  __syncthreads();
  if (threadIdx.x == 0) {
    float t = 0.0f;
#pragma unroll
    for (int i = 0; i < 8; ++i) t += waves[i];
    out[0] = t * invB;
  }
}

extern "C" void kernel_launch(void* const* d_in, const int* in_sizes, int n_in,
                              void* d_out, int out_size, void* d_ws, size_t ws_size,
                              hipStream_t stream) {
  const float* outputs = (const float*)d_in[0];
  const float* labels  = (const float*)d_in[1];
  float* out      = (float*)d_out;
  float* partials = (float*)d_ws;

  const long long nelem  = (long long)in_sizes[0];
  const long long nrows  = nelem / 10;
  const long long npairs = nrows / 2;
  const long long oddRow = (nrows & 1LL) ? (nrows - 1) : -1;

  int blocks = 2048;
  int maxBlocks = (int)(ws_size / sizeof(float));
  if (maxBlocks < 1) maxBlocks = 1;
  if (blocks > maxBlocks) blocks = maxBlocks;
  long long want = (npairs + 255) / 256;
  if (want < 1) want = 1;
  if ((long long)blocks > want) blocks = (int)want;

  wloss_partial<<<blocks, 256, 0, stream>>>(outputs, labels, partials, npairs, oddRow);

  const float invB = 1.0f / (float)nrows;   // exact for power-of-two B
  wloss_final<<<1, 256, 0, stream>>>(partials, blocks, out, invB);
}